// MLP_GCN_12257836662894
// MI455X (gfx1250) — compile-verified
//
#include <hip/hip_runtime.h>
#include <hip/hip_bf16.h>
#include <math.h>

typedef __attribute__((ext_vector_type(2))) float v2f;
typedef __attribute__((ext_vector_type(8))) float v8f;

#define HIDC 128

// ---------------- degree / normalization ----------------

__global__ void k_init_deg(float* deg, int n) {
    int i = blockIdx.x * blockDim.x + threadIdx.x;
    if (i < n) deg[i] = 1.0f;  // self-loop contributes 1
}

__global__ void k_edge_deg(const long long* __restrict__ ei, float* deg, int E) {
    int e = blockIdx.x * blockDim.x + threadIdx.x;
    if (e < E) {
        long long d = ei[(long long)E + e];  // row 1 = dst
        __hip_atomic_fetch_add(&deg[d], 1.0f, __ATOMIC_RELAXED, __HIP_MEMORY_SCOPE_AGENT);
    }
}

__global__ void k_dinv(const float* __restrict__ deg, float* dinv, int n) {
    int i = blockIdx.x * blockDim.x + threadIdx.x;
    if (i < n) {
        float d = deg[i];
        dinv[i] = d > 0.0f ? rsqrtf(d) : 0.0f;
    }
}

// ---------------- fp32 WMMA GEMM with LDS A-staging ----------------
// C = act(A[MxK] @ B[KxN] + bias). One block = one 16-row M-tile; wave w = N-tile w.
// A slab (16 x K) is staged through LDS in 128-wide K chunks, shared by all waves.
// K must be a multiple of 128 (512 and 128 here). M must be a multiple of 16.
// Lane layout per CDNA5 ISA (V_WMMA_F32_16X16X4_F32):
//   A 16x4:  lane l holds A[M=l%16][K=k+2*(l/16)+j] in vj
//   B 4x16:  lane l holds B[K=k+2*(l/16)+j][N=l%16] in vj
//   C 16x16: lane l, VGPR v holds C[M = v + 8*(l/16)][N = l%16]
template <bool GUARD>
__global__ void k_wmma_gemm_lds(const float* __restrict__ A, const float* __restrict__ B,
                                const float* __restrict__ bias, float* __restrict__ C,
                                int M, int N, int K, int relu) {
    constexpr int KC = 128;
    constexpr int LDSROW = KC + 4;            // +4 floats: rows land in distinct banks
    __shared__ float As[16 * LDSROW];         // 8448 B

    const int NT = (N + 15) >> 4;
    const int wave = threadIdx.x >> 5;
    const int lane = threadIdx.x & 31;
    const long long m0 = (long long)blockIdx.x * 16;

    const int r = lane & 15;                  // A row-in-tile / B,C column-in-tile
    const int h = lane >> 4;                  // half selector
    const int n0 = wave * 16;
    const int colg = n0 + r;
    const bool colok = GUARD ? (colg < N) : true;
    const int colc = GUARD ? (colg < N ? colg : N - 1) : colg;  // clamped, always in-bounds
    const bool active = wave < NT;            // wave-uniform

    v8f acc = {};

    for (int k0 = 0; k0 < K; k0 += KC) {
        // Cooperative, coalesced load of A[m0..m0+15][k0..k0+KC) into LDS.
        #pragma unroll
        for (int i = threadIdx.x; i < 16 * KC; i += 256) {
            int row = i >> 7;                 // KC == 128
            int col = i & (KC - 1);
            As[row * LDSROW + col] = A[(m0 + row) * (long long)K + k0 + col];
        }
        __syncthreads();

        if (active) {                         // wave-uniform: EXEC all-ones for WMMA
            const float* Ar = As + r * LDSROW + 2 * h;
            #pragma unroll 4
            for (int kk = 0; kk < KC; kk += 4) {
                v2f a, b;
                a.x = Ar[kk];
                a.y = Ar[kk + 1];
                long long kg = (long long)(k0 + kk + 2 * h);
                float bx = B[kg * N + colc];
                float by = B[(kg + 1) * N + colc];
                b.x = (GUARD && !colok) ? 0.0f : bx;   // v_cndmask, no exec change
                b.y = (GUARD && !colok) ? 0.0f : by;
                acc = __builtin_amdgcn_wmma_f32_16x16x4_f32(
                    false, a, false, b, (short)0, acc, false, false);
            }
        }
        __syncthreads();
    }

    if (active && colok) {
        float bv = bias ? bias[colc] : 0.0f;
        #pragma unroll
        for (int v = 0; v < 8; ++v) {
            float val = acc[v] + bv;
            if (relu) val = fmaxf(val, 0.0f);
            C[(m0 + v + 8 * h) * (long long)N + colg] = val;
        }
    }
}

// ---------------- GCN aggregation ----------------

// agg[n][c] = bias[c] + dinv[n]^2 * hw[n][c]   (self-loop term folded in)
__global__ void k_agg_init(const float* __restrict__ hw, const float* __restrict__ dinv,
                           const float* __restrict__ b, float* __restrict__ agg,
                           long long total) {
    long long i = (long long)blockIdx.x * blockDim.x + threadIdx.x;
    if (i >= total) return;
    int c = (int)(i & (HIDC - 1));
    long long n = i >> 7;
    float di = dinv[n];
    agg[i] = b[c] + di * di * hw[i];
}

// agg[dst][c] += dinv[src]*dinv[dst] * hw[src][c] for each real edge.
// h (51 MB) is L2-resident (192 MB L2) so random gathers/atomics stay on-chip.
__global__ void k_scatter(const long long* __restrict__ ei, const float* __restrict__ hw,
                          const float* __restrict__ dinv, float* __restrict__ agg, int E) {
    long long t = (long long)blockIdx.x * blockDim.x + threadIdx.x;
    long long e = t >> 7;  // 128 threads (channels) per edge
    int c = (int)(t & (HIDC - 1));
    if (e >= E) return;
    long long s = ei[e];
    long long d = ei[(long long)E + e];
    float nrm = dinv[s] * dinv[d];
    __hip_atomic_fetch_add(&agg[d * HIDC + c], nrm * hw[s * HIDC + c],
                           __ATOMIC_RELAXED, __HIP_MEMORY_SCOPE_AGENT);
}

__global__ void k_relu(float* x, long long total) {
    long long i = (long long)blockIdx.x * blockDim.x + threadIdx.x;
    if (i < total) x[i] = fmaxf(x[i], 0.0f);
}

// ---------------- log-softmax over 40 columns, one wave32 per row ----------------
__global__ void k_logsoftmax40(float* __restrict__ io, int M) {
    int wave = threadIdx.x >> 5;
    int lane = threadIdx.x & 31;
    int row = blockIdx.x * 8 + wave;
    if (row >= M) return;
    float* r = io + (long long)row * 40;
    float v0 = r[lane];  // lane < 32 < 40: always valid
    bool ok1 = (lane + 32) < 40;
    float v1 = ok1 ? r[lane + 32] : -INFINITY;
    float m = fmaxf(v0, v1);
    #pragma unroll
    for (int off = 16; off > 0; off >>= 1) m = fmaxf(m, __shfl_xor(m, off));
    float s = __expf(v0 - m) + (ok1 ? __expf(v1 - m) : 0.0f);
    #pragma unroll
    for (int off = 16; off > 0; off >>= 1) s += __shfl_xor(s, off);
    float lse = m + __logf(s);
    r[lane] = v0 - lse;
    if (ok1) r[lane + 32] = v1 - lse;
}

// ---------------- host driver ----------------

extern "C" void kernel_launch(void* const* d_in, const int* in_sizes, int n_in,
                              void* d_out, int out_size, void* d_ws, size_t ws_size,
                              hipStream_t stream) {
    const float* x      = (const float*)d_in[0];
    const long long* ei = (const long long*)d_in[1];  // int64 edge_index [2,E]
    const float* W1  = (const float*)d_in[2];
    const float* b1  = (const float*)d_in[3];
    const float* Wg0 = (const float*)d_in[4];
    const float* bg0 = (const float*)d_in[5];
    const float* Wg1 = (const float*)d_in[6];
    const float* bg1 = (const float*)d_in[7];
    const float* W2  = (const float*)d_in[8];
    const float* b2  = (const float*)d_in[9];
    float* out = (float*)d_out;

    const int IN_DIM = 512, OUT_DIM = 40;
    const int Nn = in_sizes[0] / IN_DIM;  // 100000 (multiple of 16)
    const int E  = in_sizes[1] / 2;       // 1,600,000

    // Workspace layout (~103 MB): deg | dinv | bufA (h) | bufB (hw)
    size_t npad = ((size_t)Nn + 63) & ~(size_t)63;
    float* deg  = (float*)d_ws;
    float* dinv = deg + npad;
    float* bufA = dinv + npad;
    float* bufB = bufA + (size_t)Nn * HIDC;

    const int T = 256;
    long long tot = (long long)Nn * HIDC;
    unsigned mblocks = (unsigned)(Nn / 16);  // 6250 M-tiles

    // Normalization
    k_init_deg<<<(Nn + T - 1) / T, T, 0, stream>>>(deg, Nn);
    k_edge_deg<<<(E + T - 1) / T, T, 0, stream>>>(ei, deg, E);
    k_dinv<<<(Nn + T - 1) / T, T, 0, stream>>>(deg, dinv, Nn);

    // h = relu(x @ W1 + b1)   [N=128: unguarded fast path]
    k_wmma_gemm_lds<false><<<mblocks, 256, 0, stream>>>(x, W1, b1, bufA, Nn, HIDC, IN_DIM, 1);

    // Two GCN conv layers
    const float* Wg[2] = {Wg0, Wg1};
    const float* bg[2] = {bg0, bg1};
    for (int L = 0; L < 2; ++L) {
        k_wmma_gemm_lds<false><<<mblocks, 256, 0, stream>>>(
            bufA, Wg[L], nullptr, bufB, Nn, HIDC, HIDC, 0);            // hw = h @ Wg
        k_agg_init<<<(unsigned)((tot + T - 1) / T), T, 0, stream>>>(   // agg = b + dinv^2*hw
            bufB, dinv, bg[L], bufA, tot);
        k_scatter<<<(unsigned)(((long long)E * HIDC + T - 1) / T), T, 0, stream>>>(
            ei, bufB, dinv, bufA, E);                                  // scatter-add edges
        k_relu<<<(unsigned)((tot + T - 1) / T), T, 0, stream>>>(bufA, tot);
    }

    // logits = h @ W2 + b2   [N=40: guarded path], then log-softmax in place
    k_wmma_gemm_lds<true><<<mblocks, 256, 0, stream>>>(bufA, W2, b2, out, Nn, OUT_DIM, HIDC, 0);
    k_logsoftmax40<<<(Nn + 7) / 8, 256, 0, stream>>>(out, Nn);
}